// STDLayer3D_LIF_47339129536610
// MI455X (gfx1250) — compile-verified
//
#include <hip/hip_runtime.h>
#include <hip/hip_bf16.h>

// ---------------------------------------------------------------------------
// STD-LIF layer, MI455X (gfx1250).
// Separable 7-tap Gaussian passes mapped onto V_WMMA_F32_16X16X4_F32:
//   Out(16x16) = A(16x24 input slab) x B(24x16 banded tap matrix), 6 K-steps.
// B is precomputed in WMMA register layout (Btab), boundary handling is
// branchless clamp+select, y/z passes use b128 vector stores/loads.
// ---------------------------------------------------------------------------

#define N1  96
#define SXY (96 * 96)        // 9216
#define CSZ (96 * 96 * 96)   // 884736 voxels per channel
#define NCH 4
#define NV  (NCH * CSZ)      // elements per 4-channel field

typedef __attribute__((ext_vector_type(2))) float v2f;
typedef __attribute__((ext_vector_type(4))) float v4f;
typedef __attribute__((ext_vector_type(8))) float v8f;

#define V8Z {0.f,0.f,0.f,0.f,0.f,0.f,0.f,0.f}

#define WMMA_F32(acc, a, b) \
  acc = __builtin_amdgcn_wmma_f32_16x16x4_f32(false, a, false, b, (short)0, acc, false, false)

// Btab layout: [set(2)][c(4)][step(6)][lane(32)][v(2)] — matches B operand
// register layout of V_WMMA_F32_16X16X4_F32 exactly.
#define BSTEP 64
#define BSET  (NCH * 6 * BSTEP)

__device__ __forceinline__ v2f bload(const float* __restrict__ p, int s) {
  return *(const v2f*)(p + s * BSTEP);
}

// branchless guarded A-element: unconditional clamped load + select-to-zero
__device__ __forceinline__ int clampj(int pj, bool inb) { return inb ? pj : 0; }

// ---------------------------------------------------------------------------
// Stage A (x-pass): out0 = Gx(u*I)[klif], out1 = Gx(u)[klif], out2 = Gx(1-2u)[k2]
// ---------------------------------------------------------------------------
__global__ void __launch_bounds__(32)
k_convA(const float* __restrict__ U, const float* __restrict__ Iimg,
        float* __restrict__ out0, float* __restrict__ out1, float* __restrict__ out2,
        const float* __restrict__ Btab)
{
  const int lane = threadIdx.x, half = lane >> 4, n = lane & 15;
  const int tileM = blockIdx.x % 6, tileJ = blockIdx.x / 6;
  const int f = blockIdx.y, c = blockIdx.z;
  const int m = tileM * 16 + n;
  const int jout0 = tileJ * 16;
  const int baseN = f * SXY;
  const int base  = c * CSZ + baseN;
  const int moffN = baseN + m * N1;     // I (no channel), x stride 1
  const int moff  = base  + m * N1;     // u
  const float* bpL = Btab + c * 6 * BSTEP + lane * 2;
  const float* bpK = bpL + BSET;

  v8f acc0 = V8Z, acc1 = V8Z, acc2 = V8Z;
#pragma unroll
  for (int s = 0; s < 6; ++s) {
    v2f aUI, aU, aW;
#pragma unroll
    for (int v = 0; v < 2; ++v) {
      const int j  = 4 * s + 2 * half + v;
      const int pj = jout0 - 3 + j;
      const bool inb = (unsigned)pj < (unsigned)N1;
      const int pjc = clampj(pj, inb);
      float uval = U[moff + pjc];
      float ival = Iimg[moffN + pjc];
      uval = inb ? uval : 0.0f;
      ival = inb ? ival : 0.0f;
      aUI[v] = uval * ival;
      aU[v]  = uval;
      aW[v]  = 1.0f - 2.0f * uval;
    }
    const v2f bL = bload(bpL, s);
    const v2f bK = bload(bpK, s);
    WMMA_F32(acc0, aUI, bL);
    WMMA_F32(acc1, aU,  bL);
    WMMA_F32(acc2, aW,  bK);
  }
#pragma unroll
  for (int v = 0; v < 8; ++v) {
    const int M = v + 8 * half;
    const int off = base + (tileM * 16 + M) * N1 + (jout0 + n);
    out0[off] = acc0[v]; out1[off] = acc1[v]; out2[off] = acc2[v];
  }
}

// ---------------------------------------------------------------------------
// y-pass, 3 fields: fields 0,1 klif; field 2 k2.  SM=1, SJ=96, SF=SXY.
// ---------------------------------------------------------------------------
__global__ void __launch_bounds__(32)
k_conv3(const float* __restrict__ in0, const float* __restrict__ in1,
        const float* __restrict__ in2,
        float* __restrict__ out0, float* __restrict__ out1, float* __restrict__ out2,
        const float* __restrict__ Btab)
{
  const int lane = threadIdx.x, half = lane >> 4, n = lane & 15;
  const int tileM = blockIdx.x % 6, tileJ = blockIdx.x / 6;
  const int f = blockIdx.y, c = blockIdx.z;
  const int m = tileM * 16 + n;
  const int jout0 = tileJ * 16;
  const int base = c * CSZ + f * SXY;
  const int moff = base + m;
  const float* bpL = Btab + c * 6 * BSTEP + lane * 2;
  const float* bpK = bpL + BSET;

  v8f acc0 = V8Z, acc1 = V8Z, acc2 = V8Z;
#pragma unroll
  for (int s = 0; s < 6; ++s) {
    v2f a0, a1, a2;
#pragma unroll
    for (int v = 0; v < 2; ++v) {
      const int j  = 4 * s + 2 * half + v;
      const int pj = jout0 - 3 + j;
      const bool inb = (unsigned)pj < (unsigned)N1;
      const int off = moff + clampj(pj, inb) * N1;
      float x0 = in0[off], x1 = in1[off], x2 = in2[off];
      a0[v] = inb ? x0 : 0.0f;
      a1[v] = inb ? x1 : 0.0f;
      a2[v] = inb ? x2 : 0.0f;
    }
    const v2f bL = bload(bpL, s);
    const v2f bK = bload(bpK, s);
    WMMA_F32(acc0, a0, bL);
    WMMA_F32(acc1, a1, bL);
    WMMA_F32(acc2, a2, bK);
  }
  const int offO = base + (tileM * 16 + 8 * half) + (jout0 + n) * N1;
  *(v4f*)(out0 + offO)     = (v4f){acc0[0], acc0[1], acc0[2], acc0[3]};
  *(v4f*)(out0 + offO + 4) = (v4f){acc0[4], acc0[5], acc0[6], acc0[7]};
  *(v4f*)(out1 + offO)     = (v4f){acc1[0], acc1[1], acc1[2], acc1[3]};
  *(v4f*)(out1 + offO + 4) = (v4f){acc1[4], acc1[5], acc1[6], acc1[7]};
  *(v4f*)(out2 + offO)     = (v4f){acc2[0], acc2[1], acc2[2], acc2[3]};
  *(v4f*)(out2 + offO + 4) = (v4f){acc2[4], acc2[5], acc2[6], acc2[7]};
}

// ---------------------------------------------------------------------------
// z-pass + Cn fuse: C1=Gz(in0), C2=Gz(in1) [klif], q=Gz(in2) [k2];
//   outCn = (C1+1e-6)/(C2+1e-6); outQ = q.   SM=1, SJ=SXY, SF=96.
// ---------------------------------------------------------------------------
__global__ void __launch_bounds__(32)
k_convZcn(const float* __restrict__ in0, const float* __restrict__ in1,
          const float* __restrict__ in2,
          float* __restrict__ outCn, float* __restrict__ outQ,
          const float* __restrict__ Btab)
{
  const int lane = threadIdx.x, half = lane >> 4, n = lane & 15;
  const int tileM = blockIdx.x % 6, tileJ = blockIdx.x / 6;
  const int f = blockIdx.y, c = blockIdx.z;
  const int m = tileM * 16 + n;
  const int jout0 = tileJ * 16;
  const int base = c * CSZ + f * N1;
  const int moff = base + m;
  const float* bpL = Btab + c * 6 * BSTEP + lane * 2;
  const float* bpK = bpL + BSET;

  v8f acc0 = V8Z, acc1 = V8Z, acc2 = V8Z;
#pragma unroll
  for (int s = 0; s < 6; ++s) {
    v2f a0, a1, a2;
#pragma unroll
    for (int v = 0; v < 2; ++v) {
      const int j  = 4 * s + 2 * half + v;
      const int pj = jout0 - 3 + j;
      const bool inb = (unsigned)pj < (unsigned)N1;
      const int off = moff + clampj(pj, inb) * SXY;
      float x0 = in0[off], x1 = in1[off], x2 = in2[off];
      a0[v] = inb ? x0 : 0.0f;
      a1[v] = inb ? x1 : 0.0f;
      a2[v] = inb ? x2 : 0.0f;
    }
    const v2f bL = bload(bpL, s);
    const v2f bK = bload(bpK, s);
    WMMA_F32(acc0, a0, bL);
    WMMA_F32(acc1, a1, bL);
    WMMA_F32(acc2, a2, bK);
  }
  const int offO = base + (tileM * 16 + 8 * half) + (jout0 + n) * SXY;
  float cn[8];
#pragma unroll
  for (int v = 0; v < 8; ++v) cn[v] = (acc0[v] + 1e-6f) / (acc1[v] + 1e-6f);
  *(v4f*)(outCn + offO)     = (v4f){cn[0], cn[1], cn[2], cn[3]};
  *(v4f*)(outCn + offO + 4) = (v4f){cn[4], cn[5], cn[6], cn[7]};
  *(v4f*)(outQ + offO)      = (v4f){acc2[0], acc2[1], acc2[2], acc2[3]};
  *(v4f*)(outQ + offO + 4)  = (v4f){acc2[4], acc2[5], acc2[6], acc2[7]};
}

// ---------------------------------------------------------------------------
// Stage D (x-pass): out0 = Gx(Cn), out1 = Gx(Cn^2)  (both klif)
// ---------------------------------------------------------------------------
__global__ void __launch_bounds__(32)
k_convXcn2(const float* __restrict__ Cn,
           float* __restrict__ out0, float* __restrict__ out1,
           const float* __restrict__ Btab)
{
  const int lane = threadIdx.x, half = lane >> 4, n = lane & 15;
  const int tileM = blockIdx.x % 6, tileJ = blockIdx.x / 6;
  const int f = blockIdx.y, c = blockIdx.z;
  const int m = tileM * 16 + n;
  const int jout0 = tileJ * 16;
  const int base = c * CSZ + f * SXY;
  const int moff = base + m * N1;
  const float* bpL = Btab + c * 6 * BSTEP + lane * 2;

  v8f acc0 = V8Z, acc1 = V8Z;
#pragma unroll
  for (int s = 0; s < 6; ++s) {
    v2f a0, a1;
#pragma unroll
    for (int v = 0; v < 2; ++v) {
      const int j  = 4 * s + 2 * half + v;
      const int pj = jout0 - 3 + j;
      const bool inb = (unsigned)pj < (unsigned)N1;
      float x = Cn[moff + clampj(pj, inb)];
      x = inb ? x : 0.0f;
      a0[v] = x; a1[v] = x * x;
    }
    const v2f bL = bload(bpL, s);
    WMMA_F32(acc0, a0, bL);
    WMMA_F32(acc1, a1, bL);
  }
#pragma unroll
  for (int v = 0; v < 8; ++v) {
    const int M = v + 8 * half;
    const int off = base + (tileM * 16 + M) * N1 + (jout0 + n);
    out0[off] = acc0[v]; out1[off] = acc1[v];
  }
}

// ---------------------------------------------------------------------------
// y-pass, 2 fields, both klif.  SM=1, SJ=96, SF=SXY.
// ---------------------------------------------------------------------------
__global__ void __launch_bounds__(32)
k_conv2(const float* __restrict__ in0, const float* __restrict__ in1,
        float* __restrict__ out0, float* __restrict__ out1,
        const float* __restrict__ Btab)
{
  const int lane = threadIdx.x, half = lane >> 4, n = lane & 15;
  const int tileM = blockIdx.x % 6, tileJ = blockIdx.x / 6;
  const int f = blockIdx.y, c = blockIdx.z;
  const int m = tileM * 16 + n;
  const int jout0 = tileJ * 16;
  const int base = c * CSZ + f * SXY;
  const int moff = base + m;
  const float* bpL = Btab + c * 6 * BSTEP + lane * 2;

  v8f acc0 = V8Z, acc1 = V8Z;
#pragma unroll
  for (int s = 0; s < 6; ++s) {
    v2f a0, a1;
#pragma unroll
    for (int v = 0; v < 2; ++v) {
      const int j  = 4 * s + 2 * half + v;
      const int pj = jout0 - 3 + j;
      const bool inb = (unsigned)pj < (unsigned)N1;
      const int off = moff + clampj(pj, inb) * N1;
      float x0 = in0[off], x1 = in1[off];
      a0[v] = inb ? x0 : 0.0f;
      a1[v] = inb ? x1 : 0.0f;
    }
    const v2f bL = bload(bpL, s);
    WMMA_F32(acc0, a0, bL);
    WMMA_F32(acc1, a1, bL);
  }
  const int offO = base + (tileM * 16 + 8 * half) + (jout0 + n) * N1;
  *(v4f*)(out0 + offO)     = (v4f){acc0[0], acc0[1], acc0[2], acc0[3]};
  *(v4f*)(out0 + offO + 4) = (v4f){acc0[4], acc0[5], acc0[6], acc0[7]};
  *(v4f*)(out1 + offO)     = (v4f){acc1[0], acc1[1], acc1[2], acc1[3]};
  *(v4f*)(out1 + offO + 4) = (v4f){acc1[4], acc1[5], acc1[6], acc1[7]};
}

// ---------------------------------------------------------------------------
// z-pass + Lif/logits fuse: Q=Gz(inQ), P=Gz(inP) [klif];
//   Lif = P - 2*I*Q + I^2 * s[x]s[y]s[z];  logits = o - mu*Lif - lam*q
// ---------------------------------------------------------------------------
__global__ void __launch_bounds__(32)
k_convZlog(const float* __restrict__ inQ, const float* __restrict__ inP,
           const float* __restrict__ q, const float* __restrict__ o,
           const float* __restrict__ Iimg, const float* __restrict__ sTab,
           const float* __restrict__ mu, const float* __restrict__ lam,
           float* __restrict__ outL, const float* __restrict__ Btab)
{
  const int lane = threadIdx.x, half = lane >> 4, n = lane & 15;
  const int tileM = blockIdx.x % 6, tileJ = blockIdx.x / 6;
  const int f = blockIdx.y, c = blockIdx.z;
  const int m = tileM * 16 + n;
  const int jout0 = tileJ * 16;
  const int base = c * CSZ + f * N1;
  const int moff = base + m;
  const float* bpL = Btab + c * 6 * BSTEP + lane * 2;

  v8f accQ = V8Z, accP = V8Z;
#pragma unroll
  for (int s = 0; s < 6; ++s) {
    v2f aQ, aP;
#pragma unroll
    for (int v = 0; v < 2; ++v) {
      const int j  = 4 * s + 2 * half + v;
      const int pj = jout0 - 3 + j;
      const bool inb = (unsigned)pj < (unsigned)N1;
      const int off = moff + clampj(pj, inb) * SXY;
      float xq = inQ[off], xp = inP[off];
      aQ[v] = inb ? xq : 0.0f;
      aP[v] = inb ? xp : 0.0f;
    }
    const v2f bL = bload(bpL, s);
    WMMA_F32(accQ, aQ, bL);
    WMMA_F32(accP, aP, bL);
  }
  const float muv = mu[0], lamv = lam[0];
  const int x0 = tileM * 16 + 8 * half;           // 8 consecutive x per lane
  const int z  = jout0 + n;
  const int offN = x0 + f * N1 + z * SXY;         // spatial (no channel)
  const int offc = c * CSZ + offN;
  const float syz = sTab[c * N1 + f] * sTab[c * N1 + z];
  const v4f Ia = *(const v4f*)(Iimg + offN);
  const v4f Ib = *(const v4f*)(Iimg + offN + 4);
  const v4f sa = *(const v4f*)(sTab + c * N1 + x0);
  const v4f sb = *(const v4f*)(sTab + c * N1 + x0 + 4);
  const v4f oa = *(const v4f*)(o + offc);
  const v4f ob = *(const v4f*)(o + offc + 4);
  const v4f qa = *(const v4f*)(q + offc);
  const v4f qb = *(const v4f*)(q + offc + 4);
  float r[8];
#pragma unroll
  for (int v = 0; v < 8; ++v) {
    const float Iv = (v < 4) ? Ia[v & 3] : Ib[v & 3];
    const float sx = (v < 4) ? sa[v & 3] : sb[v & 3];
    const float ov = (v < 4) ? oa[v & 3] : ob[v & 3];
    const float qv = (v < 4) ? qa[v & 3] : qb[v & 3];
    const float Lif = accP[v] - 2.0f * Iv * accQ[v] + Iv * Iv * (sx * syz);
    r[v] = ov - muv * Lif - lamv * qv;
  }
  *(v4f*)(outL + offc)     = (v4f){r[0], r[1], r[2], r[3]};
  *(v4f*)(outL + offc + 4) = (v4f){r[4], r[5], r[6], r[7]};
}

// ---------------------------------------------------------------------------
// Channel softmax: U[c] = softmax(X[c] / eta) over 4 channels per voxel.
// ---------------------------------------------------------------------------
__global__ void k_softmax(const float* __restrict__ X, const float* __restrict__ eta,
                          float* __restrict__ U)
{
  const int vix = blockIdx.x * blockDim.x + threadIdx.x;
  if (vix >= CSZ) return;
  const float ie = 1.0f / eta[0];
  float a0 = X[vix] * ie;
  float a1 = X[vix + CSZ] * ie;
  float a2 = X[vix + 2 * CSZ] * ie;
  float a3 = X[vix + 3 * CSZ] * ie;
  const float mx = fmaxf(fmaxf(a0, a1), fmaxf(a2, a3));
  a0 = expf(a0 - mx); a1 = expf(a1 - mx); a2 = expf(a2 - mx); a3 = expf(a3 - mx);
  const float inv = 1.0f / (a0 + a1 + a2 + a3);
  U[vix]           = a0 * inv;
  U[vix + CSZ]     = a1 * inv;
  U[vix + 2 * CSZ] = a2 * inv;
  U[vix + 3 * CSZ] = a3 * inv;
}

// ---------------------------------------------------------------------------
// Setup: normalized 1-D taps (klif<-sigma3, k2<-sigma2), boundary sums
// sTab[c][96] (conv_ones = s*s*s), and the WMMA-layout banded B table.
// ---------------------------------------------------------------------------
__global__ void k_setup(const float* __restrict__ sigma2, const float* __restrict__ sigma3,
                        float* __restrict__ klif, float* __restrict__ k2,
                        float* __restrict__ sTab, float* __restrict__ Btab)
{
  const int tid = threadIdx.x;
  if (tid < 8) {
    const int c = tid & 3;
    const bool isLif = tid >= 4;
    const float sg = isLif ? sigma3[c] : sigma2[c];
    float* dst = (isLif ? klif : k2) + c * 7;
    float tv[7];
    float sum = 0.f;
#pragma unroll
    for (int t = 0; t < 7; ++t) {
      const float d = (float)(t - 3);
      tv[t] = expf(-(d * d) / (2.0f * sg * sg));
      sum += tv[t];
    }
#pragma unroll
    for (int t = 0; t < 7; ++t) dst[t] = tv[t] / sum;
  }
  __syncthreads();
  for (int i = tid; i < NCH * N1; i += blockDim.x) {
    const int c = i / N1, p = i % N1;
    float sm = 0.f;
#pragma unroll
    for (int t = 0; t < 7; ++t) {
      const int pp = p + t - 3;
      if (pp >= 0 && pp < N1) sm += klif[c * 7 + t];
    }
    sTab[i] = sm;
  }
  // Btab[i], i = ((set*NCH + c)*6 + s)*64 + lane*2 + v
  for (int i = tid; i < 2 * BSET; i += blockDim.x) {
    const int v = i & 1, lane = (i >> 1) & 31;
    const int rest = i >> 6;
    const int s = rest % 6;
    const int sc = rest / 6;
    const int c = sc % NCH, set = sc / NCH;
    const float* tp = (set ? k2 : klif) + c * 7;
    const int half = lane >> 4, n = lane & 15;
    const int d = 4 * s + 2 * half + v - n;
    Btab[i] = ((unsigned)d <= 6u) ? tp[d] : 0.0f;
  }
}

// ---------------------------------------------------------------------------
extern "C" void kernel_launch(void* const* d_in, const int* in_sizes, int n_in,
                              void* d_out, int out_size, void* d_ws, size_t ws_size,
                              hipStream_t stream) {
  if (n_in < 7) return;
  const float* o    = (const float*)d_in[0];
  const float* Iimg = (const float*)d_in[1];
  const float* sg2  = (const float*)d_in[2];
  const float* sg3  = (const float*)d_in[3];
  const float* eta  = (const float*)d_in[4];
  const float* lam  = (const float*)d_in[5];
  const float* mu   = (const float*)d_in[6];
  float* U = (float*)d_out;   // u lives directly in the output buffer

  float* w = (float*)d_ws;
  float* t[6];
  for (int i = 0; i < 6; ++i) t[i] = w + (size_t)i * NV;
  float* klif = w + (size_t)6 * NV;   // 4*7 taps (padded to 32)
  float* k2   = klif + 32;            // 4*7 taps (padded to 32)
  float* sTab = k2 + 32;              // 4*96 boundary sums
  float* Btab = sTab + NCH * N1;      // 2*4*6*64 WMMA-layout tap matrices
  const size_t need = ((size_t)6 * NV + 64 + (size_t)NCH * N1 + 2 * BSET) * sizeof(float);
  if (ws_size < need) return;

  const dim3 cgrid(36, N1, NCH);      // 6x6 tiles, fixed coord, channel
  const dim3 cblk(32);                // one wave per 16x16 WMMA tile
  const int smblk = 256, smgrd = (CSZ + smblk - 1) / smblk;

  k_setup<<<1, 128, 0, stream>>>(sg2, sg3, klif, k2, sTab, Btab);
  k_softmax<<<smgrd, smblk, 0, stream>>>(o, eta, U);          // u0 = softmax(o/eta)

  for (int it = 0; it < 10; ++it) {
    // x-pass: u*I, u, 1-2u
    k_convA<<<cgrid, cblk, 0, stream>>>(U, Iimg, t[0], t[1], t[2], Btab);
    // y-pass (3 fields)
    k_conv3<<<cgrid, cblk, 0, stream>>>(t[0], t[1], t[2], t[3], t[4], t[5], Btab);
    // z-pass + Cn fuse: t0 = Cn, t1 = q
    k_convZcn<<<cgrid, cblk, 0, stream>>>(t[3], t[4], t[5], t[0], t[1], Btab);
    // x-pass: t2 = Gx(Cn), t3 = Gx(Cn^2)
    k_convXcn2<<<cgrid, cblk, 0, stream>>>(t[0], t[2], t[3], Btab);
    // y-pass (2 fields): t4 = Gxy(Cn), t5 = Gxy(Cn^2)
    k_conv2<<<cgrid, cblk, 0, stream>>>(t[2], t[3], t[4], t[5], Btab);
    // z-pass + Lif/logits fuse: t2 = logits
    k_convZlog<<<cgrid, cblk, 0, stream>>>(t[4], t[5], t[1], o, Iimg, sTab,
                                           mu, lam, t[2], Btab);
    // u = softmax(logits/eta)
    k_softmax<<<smgrd, smblk, 0, stream>>>(t[2], eta, U);
  }
}